// f_Cal_74543452389961
// MI455X (gfx1250) — compile-verified
//
#include <hip/hip_runtime.h>
#include <math.h>

typedef __attribute__((ext_vector_type(2))) float v2f;
typedef __attribute__((ext_vector_type(8))) float v8f;

#define NUM_SAMPLES 200
#define KDIM 100

// One wave (32 threads) per 16-sample tile. Each lane supplies the A-matrix
// elements per the documented 32-bit A 16x4 layout:
//   lanes 0-15  : row M = lane,      VGPR0 = K0, VGPR1 = K1
//   lanes 16-31 : row M = lane-16,   VGPR0 = K2, VGPR1 = K3
// B is all-ones (layout-independent), so each chained WMMA adds 4 K-elements
// to every sample's running row-sum held in the f32 accumulator.
__global__ __launch_bounds__(32)
void chi2_rowsum_wmma(const float* __restrict__ y,
                      const float* __restrict__ mu,
                      const float* __restrict__ sd,
                      const int*   __restrict__ idx,
                      float*       __restrict__ q)
{
    const int lane = threadIdx.x;       // 0..31, all active (EXEC all-1s)
    const int m    = lane & 15;         // row within tile
    const int half = lane >> 4;         // 0 -> K{0,1}, 1 -> K{2,3}
    const int s0   = blockIdx.x * 16;
    const int row  = s0 + m;

    // Branchless tail guard: clamp the gather row (loads stay in-bounds),
    // zero the contribution. No divergence around the WMMA ops.
    const int   rclamp = (row < NUM_SAMPLES) ? row : (NUM_SAMPLES - 1);
    const float valid  = (row < NUM_SAMPLES) ? 1.0f : 0.0f;
    const int   base   = rclamp * KDIM;

    v8f acc = {};                 // C starts at zero
    v2f b;  b.x = 1.0f; b.y = 1.0f;

    for (int c = 0; c < KDIM / 4; ++c) {
        const int kA = c * 4 + half * 2;
        const int i0 = idx[base + kA];
        const int i1 = idx[base + kA + 1];
        const float z0 = (y[i0] - mu[i0]) / sd[i0];
        const float z1 = (y[i1] - mu[i1]) / sd[i1];
        v2f a;
        a.x = z0 * z0 * valid;
        a.y = z1 * z1 * valid;
        // D = A x ones + C  ->  row sums accumulate, replicated across N
        acc = __builtin_amdgcn_wmma_f32_16x16x4_f32(
            false, a, false, b, (short)0, acc, false, false);
    }

    // D layout: VGPR r holds row s0+r on lanes 0-15 and row s0+8+r on 16-31.
    if (lane == 0 || lane == 16) {
        const int rbase = s0 + half * 8;
        #pragma unroll
        for (int r = 0; r < 8; ++r) {
            const int out_row = rbase + r;
            if (out_row < NUM_SAMPLES) q[out_row] = acc[r];
        }
    }
}

// Single-block reduction: mean, unbiased variance (ddof=1), KL formula.
// Double accumulation; two-pass variance sums (q - mean)^2 after the mean.
__global__ __launch_bounds__(256)
void kl_reduce(const float* __restrict__ q, float* __restrict__ out)
{
    __shared__ double red[256];
    const int t = threadIdx.x;

    const double qv = (t < NUM_SAMPLES) ? (double)q[t] : 0.0;
    red[t] = qv;
    __syncthreads();
    for (int s = 128; s > 0; s >>= 1) {
        if (t < s) red[t] += red[t + s];
        __syncthreads();
    }
    const double emp_mu = red[0] / (double)NUM_SAMPLES;
    __syncthreads();

    const double d = (t < NUM_SAMPLES) ? (qv - emp_mu) : 0.0;
    red[t] = d * d;
    __syncthreads();
    for (int s = 128; s > 0; s >>= 1) {
        if (t < s) red[t] += red[t + s];
        __syncthreads();
    }

    if (t == 0) {
        const double emp_var = red[0] / (double)(NUM_SAMPLES - 1);
        const double k     = (double)KDIM;
        const double two_k = 2.0 * k;
        const double kl = 0.5 * log(two_k / emp_var)
                        + (emp_var + (emp_mu - k) * (emp_mu - k)) / (2.0 * two_k)
                        - 0.5;
        out[0] = (float)kl;
    }
}

extern "C" void kernel_launch(void* const* d_in, const int* in_sizes, int n_in,
                              void* d_out, int out_size, void* d_ws, size_t ws_size,
                              hipStream_t stream)
{
    (void)in_sizes; (void)n_in; (void)out_size; (void)ws_size;

    const float* y   = (const float*)d_in[0];
    const float* mu  = (const float*)d_in[1];
    const float* sd  = (const float*)d_in[2];
    const int*   idx = (const int*)d_in[3];

    float* q   = (float*)d_ws;     // 200 floats of scratch
    float* out = (float*)d_out;    // single f32 KL value

    const int tiles = (NUM_SAMPLES + 15) / 16;   // 13 tiles of 16 samples
    chi2_rowsum_wmma<<<tiles, 32, 0, stream>>>(y, mu, sd, idx, q);
    kl_reduce<<<1, 256, 0, stream>>>(q, out);
}